// CounterfactualAnswerLoss_31636729103071
// MI455X (gfx1250) — compile-verified
//
#include <hip/hip_runtime.h>

typedef __attribute__((ext_vector_type(2))) float v2f;
typedef __attribute__((ext_vector_type(4))) float v4f;
typedef __attribute__((ext_vector_type(8))) float v8f;
typedef __attribute__((ext_vector_type(2))) int   v2i;
typedef __attribute__((ext_vector_type(4))) int   v4i;

#define CF_B 128
#define CF_K 16
#define CF_V 32000
#define CF_D 10
#define CF_NSPLIT 5
#define CF_VS (CF_V / CF_NSPLIT)   // 6400 V per split
#define CF_CH 256                  // V-chunk staged in LDS
#define CF_NCH (CF_VS / CF_CH)     // 25 chunks
#define CF_EPS 1e-8f

#if __has_builtin(__builtin_amdgcn_wmma_f32_16x16x4_f32)
#define HAVE_WMMA_F32_4 1
#else
#define HAVE_WMMA_F32_4 0
#endif

#if __has_builtin(__builtin_amdgcn_global_load_async_to_lds_b128) && \
    __has_builtin(__builtin_amdgcn_global_load_async_to_lds_b64)
#define HAVE_ASYNC 1
#else
#define HAVE_ASYNC 0
#endif

#if __has_builtin(__builtin_amdgcn_s_wait_asynccnt)
#define WAIT_ASYNC(n) __builtin_amdgcn_s_wait_asynccnt(n)
#else
#define WAIT_ASYNC(n) asm volatile("s_wait_asynccnt %0" ::"i"(n) : "memory")
#endif

// Typed address-space pointer casts matching the async builtin signatures
// (b128 takes int4 pointers, b64 takes int2 pointers; src=AS1, dst=AS3).
#define AS1_V4(p) ((__attribute__((address_space(1))) v4i*)(p))
#define AS3_V4(p) ((__attribute__((address_space(3))) v4i*)(p))
#define AS1_V2(p) ((__attribute__((address_space(1))) v2i*)(p))
#define AS3_V2(p) ((__attribute__((address_space(3))) v2i*)(p))

// ---------------------------------------------------------------------------
// Main kernel: per (b, split) compute partial C[16][16] = A_rows x W and
// partial per-row sums of A over the split's V-range.
// Async double-buffered LDS staging; V_WMMA_F32_16X16X4_F32 inner loop.
// ---------------------------------------------------------------------------
__global__ __launch_bounds__(256)
void cf_partial_gemm(const float* __restrict__ p_z,
                     const float* __restrict__ mix,
                     const int*   __restrict__ k_vals,
                     const float* __restrict__ coin_u,
                     const float* __restrict__ Wg,
                     float* __restrict__ wsC,     // [B][NSPLIT][16][16]
                     float* __restrict__ wsRS)    // [B][NSPLIT][16]
{
    __shared__ float sA[2][16][CF_CH + 4];        // padded rows (bank-conflict free)
    __shared__ float sWl[2][CF_CH * CF_D + 16];   // linear W slice + safety tail
    __shared__ float sRC[8][32][8];               // per-wave C fragments
    __shared__ float sRSh[8][32];                 // per-lane rowsum partials
    __shared__ float sDummy[128];                 // async sink for inactive rows

    const int b     = blockIdx.x;
    const int split = blockIdx.y;
    const int tid   = threadIdx.x;
    const int lane  = tid & 31;
    const int w     = tid >> 5;

    const int  k        = k_vals[b];
    const bool use_perm = (k >= 2) && (coin_u[b] < 0.5f);
    const float* src = (use_perm ? p_z : mix) + (size_t)b * CF_K * CF_V;

    // Pre-zero A buffers (rows >= k stay zero forever) and the W tail.
    for (int i = tid; i < 2 * 16 * (CF_CH + 4); i += 256) ((float*)sA)[i] = 0.f;
    if (tid < 16) { sWl[0][CF_CH * CF_D + tid] = 0.f; sWl[1][CF_CH * CF_D + tid] = 0.f; }
    __syncthreads();

    // Issues 9 async ops per wave: 4x b128 (A tile) + 5x b64 (W slice).
    auto issue_chunk = [&](int buf, int vbase) {
#if HAVE_ASYNC
#pragma unroll
        for (int t = 0; t < 4; ++t) {
            const int i = t * 256 + tid, row = i >> 6, c4 = i & 63;
            const float* g = src + (size_t)row * CF_V + vbase + c4 * 4;
            float* l = (row < k) ? &sA[buf][row][c4 * 4] : &sDummy[lane * 4];
            __builtin_amdgcn_global_load_async_to_lds_b128(AS1_V4(g), AS3_V4(l), 0, 0);
        }
#pragma unroll
        for (int t = 0; t < 5; ++t) {
            const int i = t * 256 + tid;
            const float* g = Wg + (size_t)vbase * CF_D + i * 2;
            float* l = &sWl[buf][i * 2];
            __builtin_amdgcn_global_load_async_to_lds_b64(AS1_V2(g), AS3_V2(l), 0, 0);
        }
#else
        (void)buf; (void)vbase;
#endif
    };
    (void)issue_chunk;

    v8f  acc = {0.f, 0.f, 0.f, 0.f, 0.f, 0.f, 0.f, 0.f};
    float rs = 0.f;
    const int Mrow = lane & 15;          // A row; also B/D column index
    const int kSel = (lane >> 4) << 1;   // 0 for lanes 0-15, 2 for lanes 16-31
    const int vStart = split * CF_VS;

#if HAVE_ASYNC
    issue_chunk(0, vStart);
#endif

    for (int c = 0; c < CF_NCH; ++c) {
        const int buf = c & 1;
#if HAVE_ASYNC
        if (c + 1 < CF_NCH) {
            issue_chunk(buf ^ 1, vStart + (c + 1) * CF_CH);
            WAIT_ASYNC(9);     // previous chunk landed; next chunk in flight
        } else {
            WAIT_ASYNC(0);
        }
        __syncthreads();
        const int bufc = buf;
#else
        const int vbase = vStart + c * CF_CH;
        __syncthreads();
#pragma unroll
        for (int t = 0; t < 4; ++t) {
            const int i = t * 256 + tid, row = i >> 6, c4 = i & 63;
            if (row < k)
                *(v4f*)&sA[0][row][c4 * 4] =
                    *(const v4f*)(src + (size_t)row * CF_V + vbase + c4 * 4);
        }
#pragma unroll
        for (int t = 0; t < 10; ++t) {
            const int i = t * 256 + tid;
            sWl[0][i] = Wg[(size_t)vbase * CF_D + i];
        }
        __syncthreads();
        const int bufc = 0;
#endif

        const int kb = w * 32;           // wave's K window within the chunk
#pragma unroll
        for (int s8 = 0; s8 < 8; ++s8) {
            const int kl = kb + s8 * 4 + kSel;
            v2f a;
            a[0] = sA[bufc][Mrow][kl];
            a[1] = sA[bufc][Mrow][kl + 1];
            rs += a[0] + a[1];
            v2f bb;
            bb[0] = sWl[bufc][kl * CF_D + Mrow];         // cols >=10 read junk,
            bb[1] = sWl[bufc][(kl + 1) * CF_D + Mrow];   // discarded at finalize
#if HAVE_WMMA_F32_4
            acc = __builtin_amdgcn_wmma_f32_16x16x4_f32(
                false, a, false, bb, (short)0, acc, false, false);
#else
            const int kbs = kb + s8 * 4;
#pragma unroll
            for (int e = 0; e < 8; ++e) {
                const int M = e + ((lane >= 16) ? 8 : 0);
                float t = acc[e];
#pragma unroll
                for (int kk = 0; kk < 4; ++kk)
                    t += sA[bufc][M][kbs + kk] * sWl[bufc][(kbs + kk) * CF_D + Mrow];
                acc[e] = t;
            }
#endif
        }
#if HAVE_ASYNC
        __syncthreads();   // buffer free before it is refilled next iteration
#endif
    }

    // ---- cross-wave reduction of C fragments and rowsums ----
#pragma unroll
    for (int e = 0; e < 8; ++e) sRC[w][lane][e] = acc[e];
    sRSh[w][lane] = rs;
    __syncthreads();

    const int M = tid >> 4, N = tid & 15;
    float cv = 0.f;
#pragma unroll
    for (int ww = 0; ww < 8; ++ww)
        cv += sRC[ww][N + ((M >= 8) ? 16 : 0)][M & 7];
    wsC[(((size_t)b * CF_NSPLIT + split) * 16 + M) * 16 + N] = cv;

    if (tid < 16) {
        float r2 = 0.f;
#pragma unroll
        for (int ww = 0; ww < 8; ++ww)
            r2 += sRSh[ww][tid] + sRSh[ww][tid + 16];
        wsRS[((size_t)b * CF_NSPLIT + split) * 16 + tid] = r2;
    }
}

// ---------------------------------------------------------------------------
// Finalize: combine splits, scale rows, softmax + JS, deterministic mean.
// ---------------------------------------------------------------------------
__device__ inline void softmax10(const float* in, float* out)
{
    float mx = in[0];
#pragma unroll
    for (int i = 1; i < CF_D; ++i) mx = fmaxf(mx, in[i]);
    float s = 0.f;
#pragma unroll
    for (int i = 0; i < CF_D; ++i) { out[i] = __expf(in[i] - mx); s += out[i]; }
    const float inv = 1.f / s;
#pragma unroll
    for (int i = 0; i < CF_D; ++i) out[i] *= inv;
}

__global__ __launch_bounds__(128)
void cf_finalize(const float* __restrict__ dlr,
                 const int*   __restrict__ k_vals,
                 const float* __restrict__ coin_u,
                 const float* __restrict__ wsC,
                 const float* __restrict__ wsRS,
                 float* __restrict__ out)
{
    const int b = threadIdx.x;   // 128 threads, one batch each
    const int  k        = k_vals[b];
    const bool use_perm = (k >= 2) && (coin_u[b] < 0.5f);

    float logits[CF_D];
#pragma unroll
    for (int n = 0; n < CF_D; ++n) logits[n] = 0.f;

    for (int M = 0; M < 16; ++M) {
        float rsum = 0.f;
#pragma unroll
        for (int s = 0; s < CF_NSPLIT; ++s)
            rsum += wsRS[((size_t)b * CF_NSPLIT + s) * 16 + M];
        const float scale = (M < k) ? (use_perm ? 1.f : 1.f / fmaxf(rsum, CF_EPS)) : 0.f;
        if (scale != 0.f) {
#pragma unroll
            for (int n = 0; n < CF_D; ++n) {
                float cv = 0.f;
#pragma unroll
                for (int s = 0; s < CF_NSPLIT; ++s)
                    cv += wsC[(((size_t)b * CF_NSPLIT + s) * 16 + M) * 16 + n];
                logits[n] += scale * cv;
            }
        }
    }
#pragma unroll
    for (int n = 0; n < CF_D; ++n) logits[n] *= (1.f / (float)CF_K);  // /K, TAU=1

    float pcf[CF_D], pref[CF_D], lref[CF_D];
    softmax10(logits, pcf);
#pragma unroll
    for (int n = 0; n < CF_D; ++n) lref[n] = dlr[(size_t)b * CF_D + n];
    softmax10(lref, pref);

    float kl_pm = 0.f, kl_qm = 0.f;
#pragma unroll
    for (int n = 0; n < CF_D; ++n) {
        const float p = fmaxf(pref[n], CF_EPS);
        const float q = fmaxf(pcf[n], CF_EPS);
        const float m = 0.5f * (p + q);
        const float lm = __logf(m);
        kl_pm += p * (__logf(p) - lm);
        kl_qm += q * (__logf(q) - lm);
    }
    const float js = 0.5f * (kl_pm + kl_qm);

    __shared__ float red[128];
    red[b] = js;
    __syncthreads();
    for (int off = 64; off > 0; off >>= 1) {
        if (b < off) red[b] += red[b + off];
        __syncthreads();
    }
    if (b == 0) out[0] = -(red[0] * (1.f / (float)CF_B));
}

// ---------------------------------------------------------------------------
extern "C" void kernel_launch(void* const* d_in, const int* in_sizes, int n_in,
                              void* d_out, int out_size, void* d_ws, size_t ws_size,
                              hipStream_t stream)
{
    (void)in_sizes; (void)n_in; (void)out_size; (void)ws_size;

    const float* p_z    = (const float*)d_in[0];   // [B,K,V]
    const float* dlr    = (const float*)d_in[1];   // [B,D]
    const int*   k_vals = (const int*)  d_in[2];   // [B]
    const float* coin_u = (const float*)d_in[3];   // [B]
    /* d_in[4] = perm_noise : unused (sum over k is permutation-invariant) */
    const float* mix    = (const float*)d_in[5];   // [B,K,V]
    const float* Wg     = (const float*)d_in[6];   // [V,D]
    float* out = (float*)d_out;

    float* wsC  = (float*)d_ws;                         // 128*5*256 floats
    float* wsRS = wsC + (size_t)CF_B * CF_NSPLIT * 256; // 128*5*16 floats

    dim3 grid(CF_B, CF_NSPLIT);
    cf_partial_gemm<<<grid, 256, 0, stream>>>(p_z, mix, k_vals, coin_u, Wg, wsC, wsRS);
    cf_finalize<<<1, 128, 0, stream>>>(dlr, k_vals, coin_u, wsC, wsRS, out);
}